// MultiHopMessagePassingLayer_52475910423180
// MI455X (gfx1250) — compile-verified
//
#include <hip/hip_runtime.h>
#include <hip/hip_bf16.h>

typedef float v2f __attribute__((ext_vector_type(2)));
typedef float v8f __attribute__((ext_vector_type(8)));

#define NHEAD 34
#define HDIM  100
#define BSZ   32
#define NNODE 200
#define MH    (NNODE * HDIM)   /* 20000 elems per (b,f) matrix */

enum GemmMode { MODE_ZH0 = 0, MODE_ZH1 = 1, MODE_SPMM = 2, MODE_ZT0 = 3 };

// ---------------------------------------------------------------------------
// Shared FP32 WMMA GEMM: C[200 x 100] = A[200 x K] @ B[K x 100] (+ epilogue).
// Block tile 128(M) x 112(N), BK = 16 (4 x v_wmma_f32_16x16x4_f32 k-steps).
// 8 waves; wave w owns rows [w*16, w*16+16), 7 v8f accumulators (112 cols).
// B tile stored K-transposed in LDS so each WMMA fragment (two consecutive
// K values per lane) is a single ds_load_2addr of adjacent floats.
// MODE_SPMM additionally computes Dh[m] = sum_k A[m,k]*dvec[k] from the same
// LDS A tiles (fused denominator matvec: A is read exactly once).
// ---------------------------------------------------------------------------
__global__ __launch_bounds__(256) void k_gemm(
    int mode,
    const float* Abase, const float* Bbase, float* Obase,
    const float* rowScaleBase,   // MODE_ZH0: start_attn
    const float* uni,            // MODE_ZH0/ZH1
    const float* dvecBase,       // MODE_SPMM
    float*       dhBase,         // MODE_SPMM
    const float* endBase,        // MODE_ZT0
    const float* dsumBase,       // MODE_ZT0
    int Kdim, int lda)
{
    __shared__ float As[128][17];     // [row][k], +1 pad: conflict-free reads
    __shared__ float Bs[112][18];     // [col][k] (transposed), +2 pad
    __shared__ float sdv[16];

    const int tid  = threadIdx.x;
    const int wid  = tid >> 5;
    const int lane = tid & 31;
    const int l15  = lane & 15;
    const int koff = (lane >> 4) << 1;          // 0 or 2 (K sub-pair per half-wave)
    const int b    = blockIdx.z;
    const int f    = blockIdx.y;
    const int mBase = blockIdx.x * 128;
    const int bf   = b * NHEAD + f;

    const float* Aop; const float* Bop; float* Op;
    const float* rsP = nullptr; const float* dvP = nullptr; float* dhP = nullptr;
    float oscale = 1.0f;

    if (mode == MODE_ZH0) {
        Aop = Abase + (size_t)b * MH;                 // X[b] (200x100)
        Bop = Bbase + (size_t)f * HDIM * HDIM;        // W[0,f]
        Op  = Obase + (size_t)bf * MH;
        rsP = rowScaleBase + (size_t)b * NNODE;       // start_attn row scaling
        oscale = uni[bf];
    } else if (mode == MODE_ZH1) {
        Aop = Abase + (size_t)bf * MH;                // Zin1[b,g] (200x100)
        Bop = Bbase + (size_t)f * HDIM * HDIM;        // W[1,g]
        Op  = Obase + (size_t)bf * MH;                // in-place over Zin1: safe
        oscale = uni[bf];
    } else if (mode == MODE_SPMM) {
        Aop = Abase + (size_t)bf * NNODE * NNODE;     // A[b,f] (200x200)
        Bop = Bbase + (size_t)bf * MH;                // Zh[b,f]
        Op  = Obase + (size_t)bf * MH;
        dvP = dvecBase + (size_t)bf * NNODE;
        dhP = dhBase   + (size_t)bf * NNODE;
    } else {                                          // MODE_ZT0
        Aop = Abase + (size_t)b * MH;                 // S[b] (200x100)
        Bop = Bbase;                                  // Wpad0 = Wi[1]
        Op  = Obase + (size_t)b * MH;                 // -> d_out[0]
    }

    v8f acc[7];
#pragma unroll
    for (int j = 0; j < 7; ++j)
#pragma unroll
        for (int r = 0; r < 8; ++r) acc[j][r] = 0.0f;
    float dacc = 0.0f;

    for (int k0 = 0; k0 < Kdim; k0 += 16) {
        // ---- stage A tile 128x16 (8 elems/thread, coalesced along K) ----
        {
            int row  = tid >> 1;
            int kg   = (tid & 1) * 8;
            int grow = mBase + row;
            bool rv  = grow < NNODE;
            float rs = 1.0f;
            if (rsP && rv) rs = rsP[grow];
            const float* ap = Aop + (size_t)grow * lda + (k0 + kg);
#pragma unroll
            for (int j = 0; j < 8; ++j) {
                float v = 0.0f;
                if (rv && (k0 + kg + j) < Kdim) v = ap[j];
                As[row][kg + j] = v * rs;
            }
        }
        // ---- stage B tile 16x112 -> LDS transposed [col][k] ----
        {
            int kr = tid >> 4;                        // 0..15
            int cg = (tid & 15) * 7;                  // 0,7,...,105
            bool kv = (k0 + kr) < Kdim;
            const float* bp = Bop + (size_t)(k0 + kr) * HDIM + cg;
#pragma unroll
            for (int j = 0; j < 7; ++j) {
                int c = cg + j;
                float v = 0.0f;
                if (kv && c < HDIM) v = bp[j];
                Bs[c][kr] = v;
            }
        }
        if (mode == MODE_SPMM && tid >= 240) {
            int kk = k0 + (tid - 240);
            sdv[tid - 240] = (kk < Kdim) ? dvP[kk] : 0.0f;
        }
        __syncthreads();

        // ---- 4 k-steps x 7 col-tiles of v_wmma_f32_16x16x4_f32 ----
#pragma unroll
        for (int kk = 0; kk < 4; ++kk) {
            v2f a;
            a.x = As[wid * 16 + l15][kk * 4 + koff];
            a.y = As[wid * 16 + l15][kk * 4 + koff + 1];
#pragma unroll
            for (int j = 0; j < 7; ++j) {
                v2f bb;
                bb.x = Bs[j * 16 + l15][kk * 4 + koff];
                bb.y = Bs[j * 16 + l15][kk * 4 + koff + 1];
                acc[j] = __builtin_amdgcn_wmma_f32_16x16x4_f32(
                    false, a, false, bb, (short)0, acc[j], false, false);
            }
        }
        // ---- fused denominator matvec (SPMM): Dh[m] += A[m,k]*dvec[k] ----
        if (mode == MODE_SPMM && tid < 128) {
#pragma unroll
            for (int k = 0; k < 16; ++k) dacc += As[tid][k] * sdv[k];
        }
        __syncthreads();
    }

    // ---- epilogue: C/D layout — VGPR r: lanes<16 -> M=r, lanes>=16 -> M=r+8
    const int mrow0 = mBase + wid * 16 + ((lane >> 4) << 3);
#pragma unroll
    for (int j = 0; j < 7; ++j) {
        int n = j * 16 + l15;
        if (n < HDIM) {
#pragma unroll
            for (int r = 0; r < 8; ++r) {
                int m = mrow0 + r;
                if (m < NNODE) {
                    float v = acc[j][r];
                    if (mode == MODE_ZT0) {
                        float e = endBase[(size_t)b * NNODE + m];
                        v = v * e / (e * dsumBase[(size_t)b * NNODE + m] + 1e-20f);
                    } else {
                        v *= oscale;
                    }
                    Op[(size_t)m * HDIM + n] = v;
                }
            }
        }
    }
    if (mode == MODE_SPMM && tid < 128) {
        int m = mBase + tid;
        if (m < NNODE) dhP[m] = dacc;
    }
}

// ---------------------------------------------------------------------------
// Small elementwise / reduction helpers
// ---------------------------------------------------------------------------
__global__ void k_dvec0(const float* start, const float* uni, float* dvec) {
    int idx = blockIdx.x * blockDim.x + threadIdx.x;       // b*NHEAD*NNODE
    if (idx >= BSZ * NHEAD * NNODE) return;
    int n = idx % NNODE;
    int f = (idx / NNODE) % NHEAD;
    int b = idx / (NHEAD * NNODE);
    dvec[idx] = start[b * NNODE + n] * uni[b * NHEAD + f];
}

// Zin1[b,g,m] = sum_f ZhA0[b,f,m] * trans[b,f,g]   (m = n*H+i flat)
__global__ __launch_bounds__(256) void k_transmix(const float* ZhA,
                                                  const float* trans,
                                                  float* Zin) {
    __shared__ float tr[NHEAD * NHEAD];
    int b = blockIdx.y;
    for (int i = threadIdx.x; i < NHEAD * NHEAD; i += 256)
        tr[i] = trans[(size_t)b * NHEAD * NHEAD + i];
    __syncthreads();
    int m = blockIdx.x * 256 + threadIdx.x;
    if (m >= MH) return;
    float acc[NHEAD];
#pragma unroll
    for (int g = 0; g < NHEAD; ++g) acc[g] = 0.0f;
    for (int f = 0; f < NHEAD; ++f) {
        float v = ZhA[((size_t)(b * NHEAD + f)) * MH + m];
#pragma unroll
        for (int g = 0; g < NHEAD; ++g) acc[g] += v * tr[f * NHEAD + g];
    }
    for (int g = 0; g < NHEAD; ++g)
        Zin[((size_t)(b * NHEAD + g)) * MH + m] = acc[g];
}

// dvec1[b,g,n] = uni[b,g] * sum_f Dh0[b,f,n]*trans[b,f,g]
__global__ void k_dvec1(const float* Dh0, const float* trans, const float* uni,
                        float* dvec) {
    int idx = blockIdx.x * blockDim.x + threadIdx.x;
    if (idx >= BSZ * NHEAD * NNODE) return;
    int n = idx % NNODE;
    int g = (idx / NNODE) % NHEAD;
    int b = idx / (NHEAD * NNODE);
    float s = 0.0f;
    for (int f = 0; f < NHEAD; ++f)
        s += Dh0[((size_t)(b * NHEAD + f)) * NNODE + n] *
             trans[(size_t)b * NHEAD * NHEAD + f * NHEAD + g];
    dvec[idx] = uni[b * NHEAD + g] * s;
}

// S[b,m] = sum_f ZhA0[b,f,m]
__global__ void k_reduceS(const float* ZhA, float* S) {
    int idx = blockIdx.x * blockDim.x + threadIdx.x;
    if (idx >= BSZ * MH) return;
    int b = idx / MH, m = idx % MH;
    float s = 0.0f;
    for (int f = 0; f < NHEAD; ++f)
        s += ZhA[((size_t)(b * NHEAD + f)) * MH + m];
    S[idx] = s;
}

// Dsum[b,n] = sum_f Dh[b,f,n]
__global__ void k_dsum(const float* Dh, float* Dsum) {
    int idx = blockIdx.x * blockDim.x + threadIdx.x;
    if (idx >= BSZ * NNODE) return;
    int b = idx / NNODE, n = idx % NNODE;
    float s = 0.0f;
    for (int f = 0; f < NHEAD; ++f)
        s += Dh[((size_t)(b * NHEAD + f)) * NNODE + n];
    Dsum[idx] = s;
}

// out1[b,m] = end[b,n]*sum_f ZhA1[b,f,m] / (end[b,n]*D1sum[b,n] + eps)
__global__ void k_out1(const float* ZhA1, const float* endA, const float* D1s,
                       float* out) {
    int idx = blockIdx.x * blockDim.x + threadIdx.x;
    if (idx >= BSZ * MH) return;
    int b = idx / MH, m = idx % MH, n = m / HDIM;
    float s = 0.0f;
    for (int f = 0; f < NHEAD; ++f)
        s += ZhA1[((size_t)(b * NHEAD + f)) * MH + m];
    float e = endA[b * NNODE + n];
    out[(size_t)BSZ * MH + idx] = e * s / (e * D1s[b * NNODE + n] + 1e-20f);
}

// ---------------------------------------------------------------------------
extern "C" void kernel_launch(void* const* d_in, const int* in_sizes, int n_in,
                              void* d_out, int out_size, void* d_ws, size_t ws_size,
                              hipStream_t stream) {
    const float* X     = (const float*)d_in[0];
    const float* Amat  = (const float*)d_in[1];
    const float* start = (const float*)d_in[2];
    const float* endA  = (const float*)d_in[3];
    const float* uni   = (const float*)d_in[4];
    const float* trans = (const float*)d_in[5];
    const float* w_vs  = (const float*)d_in[6];
    float* out = (float*)d_out;
    float* ws  = (float*)d_ws;

    const size_t BIG = (size_t)BSZ * NHEAD * MH;         // 21,760,000
    float* ZH    = ws;                                   // Zh0 / Zin1 / Zh1
    float* ZHA   = ws + BIG;                             // ZhA0 / ZhA1
    float* Sbuf  = ws + 2 * BIG;                         // 640,000
    float* DVEC0 = Sbuf  + (size_t)BSZ * MH;
    float* DH0   = DVEC0 + (size_t)BSZ * NHEAD * NNODE;
    float* DVEC1 = DH0   + (size_t)BSZ * NHEAD * NNODE;
    float* DH1   = DVEC1 + (size_t)BSZ * NHEAD * NNODE;
    float* D0S   = DH1   + (size_t)BSZ * NHEAD * NNODE;
    float* D1S   = D0S   + (size_t)BSZ * NNODE;

    const float* W0    = w_vs;                           // w_vs[0, f]
    const float* W1    = w_vs + (size_t)35 * HDIM * HDIM;// w_vs[1, g]
    const float* WPAD0 = w_vs + (size_t)(1 * 35 + 34) * HDIM * HDIM; // Wi[1]

    dim3 gFull(2, NHEAD, BSZ), gB(2, 1, BSZ);
    int nBFN = BSZ * NHEAD * NNODE;                      // 217,600
    int nBM  = BSZ * MH;                                 // 640,000

    // t = 0
    k_gemm<<<gFull, 256, 0, stream>>>(MODE_ZH0, X, W0, ZH, start, uni,
                                      nullptr, nullptr, nullptr, nullptr,
                                      HDIM, HDIM);
    k_dvec0<<<(nBFN + 255) / 256, 256, 0, stream>>>(start, uni, DVEC0);
    k_gemm<<<gFull, 256, 0, stream>>>(MODE_SPMM, Amat, ZH, ZHA, nullptr, nullptr,
                                      DVEC0, DH0, nullptr, nullptr,
                                      NNODE, NNODE);
    k_reduceS<<<(nBM + 255) / 256, 256, 0, stream>>>(ZHA, Sbuf);
    k_dsum<<<(BSZ * NNODE + 255) / 256, 256, 0, stream>>>(DH0, D0S);
    k_transmix<<<dim3((MH + 255) / 256, BSZ), 256, 0, stream>>>(ZHA, trans, ZH);
    k_dvec1<<<(nBFN + 255) / 256, 256, 0, stream>>>(DH0, trans, uni, DVEC1);
    k_gemm<<<gB, 256, 0, stream>>>(MODE_ZT0, Sbuf, WPAD0, out, nullptr, nullptr,
                                   nullptr, nullptr, endA, D0S,
                                   HDIM, HDIM);
    // t = 1
    k_gemm<<<gFull, 256, 0, stream>>>(MODE_ZH1, ZH, W1, ZH, nullptr, uni,
                                      nullptr, nullptr, nullptr, nullptr,
                                      HDIM, HDIM);
    k_gemm<<<gFull, 256, 0, stream>>>(MODE_SPMM, Amat, ZH, ZHA, nullptr, nullptr,
                                      DVEC1, DH1, nullptr, nullptr,
                                      NNODE, NNODE);
    k_dsum<<<(BSZ * NNODE + 255) / 256, 256, 0, stream>>>(DH1, D1S);
    k_out1<<<(nBM + 255) / 256, 256, 0, stream>>>(ZHA, endA, D1S, out);
}